// OrthogonalLayer1D_46282567582179
// MI455X (gfx1250) — compile-verified
//
#include <hip/hip_runtime.h>

// CholeskyQR for batched tall-skinny QR on gfx1250 (wave32, WMMA, async-LDS).
// For each sample s: X = x[:, s, :] (128 x 1024, rows contiguous)
//   G = X X^T            (WMMA f32 16x16x4 SYRK, K=1024)
//   G = L L^T            (in-LDS Cholesky, diag > 0 => sign canonical)
//   Li = L^-1            (per-column forward substitution)
//   out[:, s, :] = Li X  (WMMA f32 16x16x4 GEMM, K=128)
// Global->LDS staging uses GLOBAL_LOAD_ASYNC_TO_LDS_B128 with a double-
// buffered chunk pipeline (ASYNCcnt), hiding HBM latency behind WMMA bursts.

typedef __attribute__((ext_vector_type(2))) float v2f;
typedef __attribute__((ext_vector_type(8))) float v8f;

#define NM 128        // n_models (panel width)
#define NS 2048       // n_samples
#define DIM 1024      // panel height
#define KC 64         // streamed chunk width (K in phase 1, D in phase 3)
#define NCHUNK (DIM / KC)
#define XS_STRIDE 68  // 64 + 4 pad: 16B-aligned rows, bank-skewed
#define G_STRIDE 132  // 128 + 4 pad

__device__ __forceinline__ void wait_async0() {
#if __has_builtin(__builtin_amdgcn_s_wait_asynccnt)
    __builtin_amdgcn_s_wait_asynccnt(0);
#else
    asm volatile("s_wait_asynccnt 0" ::: "memory");
#endif
}

// Issue async copies of one 128x64 f32 chunk (32 KB) into an LDS buffer.
// 2048 x 16B transfers; 256 threads x 8 issues; per-wave ASYNCcnt += 8.
__device__ __forceinline__ void stage_chunk_async(const float* __restrict__ x,
                                                  float* lbuf, int s, int kc,
                                                  int tid) {
    #pragma unroll
    for (int i = 0; i < 8; ++i) {
        const int t   = tid + i * 256;
        const int row = t >> 4;           // 0..127
        const int c4  = (t & 15) << 2;    // 0,4,...,60
        const float*   g   = x + ((size_t)row * NS + s) * DIM + kc + c4;
        const unsigned lds = (unsigned)(uintptr_t)(lbuf + row * XS_STRIDE + c4);
        asm volatile("global_load_async_to_lds_b128 %0, %1, off"
                     :: "v"(lds), "v"(g) : "memory");
    }
}

__launch_bounds__(256, 1)
__global__ void OrthogonalLayer1D_cholqr_kernel(const float* __restrict__ x,
                                                float* __restrict__ out) {
    __shared__ float Xs[2][NM * XS_STRIDE];  // double-buffered 128x64 chunks
    __shared__ float G [NM * G_STRIDE];      // Gram matrix -> Cholesky L
    __shared__ float Li[NM * G_STRIDE];      // L^-1 (lower triangular)

    const int s    = blockIdx.x;
    const int tid  = threadIdx.x;
    const int lane = tid & 31;
    const int wave = tid >> 5;      // 0..7
    const int lrow = lane & 15;     // M/N index within fragment
    const int lhi  = lane >> 4;     // K-half select

    // ---------------- Phase 1: G = X X^T (double-buffered SYRK) -----------
    v8f acc[8];
    #pragma unroll
    for (int j = 0; j < 8; ++j)
        #pragma unroll
        for (int e = 0; e < 8; ++e) acc[j][e] = 0.0f;

    stage_chunk_async(x, Xs[0], s, 0, tid);

    for (int c = 0; c < NCHUNK; ++c) {
        wait_async0();        // my async writes into Xs[c&1] done
        __syncthreads();      // everyone's writes done; prev chunk consumed
        if (c + 1 < NCHUNK)
            stage_chunk_async(x, Xs[(c + 1) & 1], s, (c + 1) * KC, tid);
        const float* cur = Xs[c & 1];

        #pragma unroll
        for (int k = 0; k < KC; k += 4) {
            const int kk = k + 2 * lhi;
            v2f a;
            a.x = cur[(wave * 16 + lrow) * XS_STRIDE + kk];
            a.y = cur[(wave * 16 + lrow) * XS_STRIDE + kk + 1];
            #pragma unroll
            for (int j = 0; j < 8; ++j) {
                v2f b;
                b.x = cur[(j * 16 + lrow) * XS_STRIDE + kk];
                b.y = cur[(j * 16 + lrow) * XS_STRIDE + kk + 1];
                acc[j] = __builtin_amdgcn_wmma_f32_16x16x4_f32(
                    false, a, false, b, (short)0, acc[j], false, false);
            }
        }
    }

    // Prefetch phase-3 chunk 0 now: overlaps Cholesky + triangular inverse.
    stage_chunk_async(x, Xs[0], s, 0, tid);

    // spill accumulators -> G (C/D layout: row = r + 8*lhi, col = lane&15)
    #pragma unroll
    for (int j = 0; j < 8; ++j)
        #pragma unroll
        for (int r = 0; r < 8; ++r)
            G[(wave * 16 + r + 8 * lhi) * G_STRIDE + j * 16 + lrow] = acc[j][r];
    __syncthreads();

    // ---------------- Phase 2: Cholesky G = L L^T (in place, lower) --------
    for (int k = 0; k < NM; ++k) {
        if (tid == 0) G[k * G_STRIDE + k] = sqrtf(G[k * G_STRIDE + k]);
        __syncthreads();
        const float inv = 1.0f / G[k * G_STRIDE + k];
        for (int i = k + 1 + tid; i < NM; i += 256)
            G[i * G_STRIDE + k] *= inv;
        __syncthreads();
        for (int i = k + 1 + tid; i < NM; i += 256) {
            const float lik = G[i * G_STRIDE + k];
            for (int j = k + 1; j <= i; ++j)
                G[i * G_STRIDE + j] -= lik * G[j * G_STRIDE + k];
        }
        __syncthreads();
    }

    // ---------------- Phase 3a: Li = L^-1 (one column per thread) ----------
    if (tid < NM) {
        const int col = tid;
        for (int i = 0; i < col; ++i) Li[i * G_STRIDE + col] = 0.0f;
        for (int i = col; i < NM; ++i) {
            float sum = (i == col) ? 1.0f : 0.0f;
            for (int j = col; j < i; ++j)
                sum -= G[i * G_STRIDE + j] * Li[j * G_STRIDE + col];
            Li[i * G_STRIDE + col] = sum / G[i * G_STRIDE + i];
        }
    }
    __syncthreads();

    // ---------------- Phase 3b: out[:, s, :] = Li * X (double-buffered) ----
    for (int c = 0; c < NCHUNK; ++c) {
        wait_async0();
        __syncthreads();
        if (c + 1 < NCHUNK)
            stage_chunk_async(x, Xs[(c + 1) & 1], s, (c + 1) * KC, tid);
        const float* cur = Xs[c & 1];
        const int dc = c * KC;

        #pragma unroll
        for (int nt = 0; nt < 4; ++nt) {
            v8f cc;
            #pragma unroll
            for (int e = 0; e < 8; ++e) cc[e] = 0.0f;

            #pragma unroll
            for (int k = 0; k < NM; k += 4) {
                const int kk = k + 2 * lhi;
                v2f a;
                a.x = Li[(wave * 16 + lrow) * G_STRIDE + kk];
                a.y = Li[(wave * 16 + lrow) * G_STRIDE + kk + 1];
                v2f b;  // B[kb][n] = cur[k+kb][nt*16+n]
                b.x = cur[(kk)     * XS_STRIDE + nt * 16 + lrow];
                b.y = cur[(kk + 1) * XS_STRIDE + nt * 16 + lrow];
                cc = __builtin_amdgcn_wmma_f32_16x16x4_f32(
                    false, a, false, b, (short)0, cc, false, false);
            }

            // store tile: M = wave*16 + r + 8*lhi, d = dc + nt*16 + lrow
            #pragma unroll
            for (int r = 0; r < 8; ++r) {
                const int m = wave * 16 + r + 8 * lhi;
                out[((size_t)m * NS + s) * DIM + dc + nt * 16 + lrow] = cc[r];
            }
        }
    }
}

extern "C" void kernel_launch(void* const* d_in, const int* in_sizes, int n_in,
                              void* d_out, int out_size, void* d_ws, size_t ws_size,
                              hipStream_t stream) {
    const float* x = (const float*)d_in[0];
    float* out = (float*)d_out;
    OrthogonalLayer1D_cholqr_kernel<<<NS, 256, 0, stream>>>(x, out);
}